// DirectGridVolume_61993557950729
// MI455X (gfx1250) — compile-verified
//
#include <hip/hip_runtime.h>
#include <cmath>

typedef __attribute__((ext_vector_type(2))) float v2f;
typedef __attribute__((ext_vector_type(8))) float v8f;

namespace {
constexpr float kC0  = 0.28209479177387814f;
constexpr float kC1  = 0.4886025119029199f;
constexpr float kC20 = 1.0925484305920792f;
constexpr float kC21 = -1.0925484305920792f;
constexpr float kC22 = 0.31539156525252005f;
constexpr float kC23 = -1.0925484305920792f;
constexpr float kC24 = 0.5462742152960396f;
}

// select element i (0..7) of a v8f without memory round-trip
__device__ __forceinline__ float pick8(v8f d, int i) {
  float r = d[0];
  r = (i == 1) ? d[1] : r;
  r = (i == 2) ? d[2] : r;
  r = (i == 3) ? d[3] : r;
  r = (i == 4) ? d[4] : r;
  r = (i == 5) ? d[5] : r;
  r = (i == 6) ? d[6] : r;
  r = (i == 7) ? d[7] : r;
  return r;
}

__global__ __launch_bounds__(256) void dgv_kernel(
    const float* __restrict__ xyz,
    const float* __restrict__ vdirs,
    const float* __restrict__ dgrid,
    const float* __restrict__ shgrid,
    float* __restrict__ out,
    int N, int R)
{
  const int  lane   = (int)(threadIdx.x & 31u);
  const int  gid    = blockIdx.x * blockDim.x + threadIdx.x;
  const bool active = gid < N;
  const int  n      = active ? gid : (N - 1);  // clamp: EXEC stays all-ones for WMMA

  const int  R2 = R * R;
  const long R3 = (long)R2 * R;

  // ---- trilinear setup (grid_sample, align_corners=True, zeros padding) ----
  float px = xyz[3 * n + 0] / 1.5f;
  float py = xyz[3 * n + 1] / 1.5f;
  float pz = xyz[3 * n + 2] / 1.5f;
  float fx = (px + 1.0f) * 0.5f * (float)(R - 1);
  float fy = (py + 1.0f) * 0.5f * (float)(R - 1);
  float fz = (pz + 1.0f) * 0.5f * (float)(R - 1);
  float x0f = floorf(fx), y0f = floorf(fy), z0f = floorf(fz);
  float wx1 = fx - x0f, wy1 = fy - y0f, wz1 = fz - z0f;
  int x0 = (int)x0f, y0 = (int)y0f, z0 = (int)z0f;

  int   coff[8];
  float cw[8];
#pragma unroll
  for (int dz = 0; dz < 2; ++dz) {
#pragma unroll
    for (int dy = 0; dy < 2; ++dy) {
#pragma unroll
      for (int dx = 0; dx < 2; ++dx) {
        int xi = x0 + dx, yi = y0 + dy, zi = z0 + dz;
        bool inb = (xi >= 0) && (xi < R) && (yi >= 0) && (yi < R) &&
                   (zi >= 0) && (zi < R);
        int xc = min(max(xi, 0), R - 1);
        int yc = min(max(yi, 0), R - 1);
        int zc = min(max(zi, 0), R - 1);
        int j  = dz * 4 + dy * 2 + dx;
        coff[j] = zc * R2 + yc * R + xc;
        float wx = dx ? wx1 : (1.0f - wx1);
        float wy = dy ? wy1 : (1.0f - wy1);
        float wz = dz ? wz1 : (1.0f - wz1);
        cw[j] = wx * wy * wz * (inb ? 1.0f : 0.0f);
      }
    }
  }

  // ---- gather: density (1 ch) + SH (27 ch), 8 corners each ----
  float dens = 0.0f;
#pragma unroll
  for (int j = 0; j < 8; ++j) dens += cw[j] * dgrid[coff[j]];

  float shf[27];
#pragma unroll
  for (int c = 0; c < 27; ++c) {
    const float* base = shgrid + (long)c * R3;
    float a = 0.0f;
#pragma unroll
    for (int j = 0; j < 8; ++j) a += cw[j] * base[coff[j]];
    shf[c] = a;
  }

  // ---- SH basis ----
  float vx = vdirs[3 * n + 0], vy = vdirs[3 * n + 1], vz = vdirs[3 * n + 2];
  float basis[9];
  basis[0] = kC0;
  basis[1] = -kC1 * vy;
  basis[2] = kC1 * vz;
  basis[3] = -kC1 * vx;
  basis[4] = kC20 * vx * vy;
  basis[5] = kC21 * vy * vz;
  basis[6] = kC22 * (2.0f * vz * vz - vx * vx - vy * vy);
  basis[7] = kC23 * vx * vz;
  basis[8] = kC24 * (vx * vx - vy * vy);

  // ---- einsum via V_WMMA_F32_16X16X4_F32 rank-1 updates ----
  // Two 16-point tiles per wave32. One-hot K=0: A[m,0]=basis_m[k] (VGPR0,
  // lanes 0-15, M=lane), B[0,n]=sh_n[c][k] (VGPR0, lanes 0-15, N=lane);
  // after 9 chained WMMAs, D[n,n] = dot(basis_n, sh_n[c]).
  const bool lo    = lane < 16;
  const int  hiSrc = (lane & 15) + 16;              // pull tile1 data down
  float basisHi[9];
#pragma unroll
  for (int k = 0; k < 9; ++k) basisHi[k] = __shfl(basis[k], hiSrc, 32);

  const int m7 = lane & 7;
  const int m15 = lane & 15;
  const int diagSrc = (m15 < 8) ? m15 : (m15 + 16); // diag lives at lanes 0-7 / 24-31

  float rgb[3];
#pragma unroll
  for (int c = 0; c < 3; ++c) {
    v8f d0 = {};
    v8f d1 = {};
#pragma unroll
    for (int k = 0; k < 9; ++k) {
      float bLo = shf[c * 9 + k];
      float bHi = __shfl(bLo, hiSrc, 32);
      v2f a0; a0[0] = lo ? basis[k]   : 0.0f; a0[1] = 0.0f;
      v2f b0; b0[0] = lo ? bLo        : 0.0f; b0[1] = 0.0f;
      v2f a1; a1[0] = lo ? basisHi[k] : 0.0f; a1[1] = 0.0f;
      v2f b1; b1[0] = lo ? bHi        : 0.0f; b1[1] = 0.0f;
      d0 = __builtin_amdgcn_wmma_f32_16x16x4_f32(false, a0, false, b0,
                                                 (short)0, d0, false, false);
      d1 = __builtin_amdgcn_wmma_f32_16x16x4_f32(false, a1, false, b1,
                                                 (short)0, d1, false, false);
    }
    // diag(n): n<8 -> VGPR n @ lane n ; n>=8 -> VGPR n-8 @ lane n+16
    float sel0 = pick8(d0, m7);
    float sel1 = pick8(d1, m7);
    float r0 = __shfl(sel0, diagSrc, 32);
    float r1 = __shfl(sel1, diagSrc, 32);
    float dot = lo ? r0 : r1;
    rgb[c] = 1.0f / (1.0f + expf(-dot));
  }

  if (active) {
    out[gid] = fmaxf(dens, 0.0f);
    long rb = (long)N + 3L * (long)gid;
    out[rb + 0] = rgb[0];
    out[rb + 1] = rgb[1];
    out[rb + 2] = rgb[2];
  }
}

extern "C" void kernel_launch(void* const* d_in, const int* in_sizes, int n_in,
                              void* d_out, int out_size, void* d_ws, size_t ws_size,
                              hipStream_t stream) {
  const float* xyz = (const float*)d_in[0];
  const float* vd  = (const float*)d_in[1];
  const float* dg  = (const float*)d_in[2];
  const float* sg  = (const float*)d_in[3];
  float* out = (float*)d_out;

  int N = in_sizes[0] / 3;
  if (N <= 0) return;

  long dsz = (long)in_sizes[2];                 // density grid = R^3
  int R = (int)lround(cbrt((double)dsz));
  while ((long)R * R * R < dsz) ++R;
  while (R > 1 && (long)R * R * R > dsz) --R;

  dim3 block(256);
  dim3 grid((unsigned)((N + 255) / 256));
  hipLaunchKernelGGL(dgv_kernel, grid, block, 0, stream,
                     xyz, vd, dg, sg, out, N, R);
}